// SparseGraphLearn_90915867721728
// MI455X (gfx1250) — compile-verified
//
#include <hip/hip_runtime.h>
#include <hip/hip_bf16.h>

typedef __attribute__((ext_vector_type(2))) float v2f;
typedef __attribute__((ext_vector_type(8))) float v8f;

#define GAT_N      50000
#define GAT_IN     512
#define GAT_OUT    128
#define GAT_DEG    16
#define XS_STRIDE  516   // 512 + 4 pad: A-frag lane l starts at bank 4*l -> conflict-free
#define WS_STRIDE  136   // 128 + 8 pad: B-frag k rows shift banks by 8k; half-waves disjoint
#define KPANEL     128   // K-panel of W staged in LDS per iteration

// -------------------------------------------------------------------------
// Kernel 1: h = x @ W   (fp32, V_WMMA_F32_16X16X4_F32)
// Workgroup = 16 rows x 128 cols, 8 waves (one 16x16 col tile each).
// x tile (16x512) staged in LDS once; W staged in LDS as 128x128 K-panels
// shared by all 8 waves -> ~0.56 global loads per WMMA instead of 2.
// -------------------------------------------------------------------------
__global__ __launch_bounds__(256) void sgl_gemm_h_f32wmma(
    const float* __restrict__ x, const float* __restrict__ W,
    float* __restrict__ h) {
  __shared__ float xs[16 * XS_STRIDE];       //  33,024 B
  __shared__ float ws[KPANEL * WS_STRIDE];   //  69,632 B  (total ~103 KB)

  const int tid  = threadIdx.x;
  const int row0 = blockIdx.x * 16;

  // ---- cooperative load of the 16x512 x tile (2048 float4 / 256 thr = 8 ea)
  const float4* xg = reinterpret_cast<const float4*>(x + (size_t)row0 * GAT_IN);
#pragma unroll
  for (int i = 0; i < 8; ++i) {
    int f  = tid + i * 256;
    int r  = f >> 7;          // f / 128
    int cc = f & 127;         // f % 128
    float4 v = xg[r * (GAT_IN / 4) + cc];
    *reinterpret_cast<float4*>(&xs[r * XS_STRIDE + cc * 4]) = v;
  }

  const int wave   = tid >> 5;         // 0..7 -> column tile
  const int lane   = tid & 31;
  const int lane16 = lane & 15;
  const int khalf  = (lane >> 4) * 2;  // lanes 0-15: K=0,1 ; lanes 16-31: K=2,3
  const int col0   = wave * 16;

  const float* ap  = &xs[lane16 * XS_STRIDE + khalf];       // A frags (LDS)
  const float* wsp = &ws[khalf * WS_STRIDE + col0 + lane16]; // B frags (LDS)

  v8f c = {};
  for (int kk = 0; kk < GAT_IN; kk += KPANEL) {
    __syncthreads();  // xs ready (1st iter) / previous panel fully consumed

    // ---- cooperative load of W[kk..kk+128) x 128 (4096 float4 / 256 = 16 ea)
    const float4* Wg = reinterpret_cast<const float4*>(W + (size_t)kk * GAT_OUT);
#pragma unroll
    for (int i = 0; i < 16; ++i) {
      int f  = tid + i * 256;
      int r  = f >> 5;        // f / 32  (32 float4 per 128-col row)
      int cc = f & 31;
      float4 v = Wg[r * (GAT_OUT / 4) + cc];
      *reinterpret_cast<float4*>(&ws[r * WS_STRIDE + cc * 4]) = v;
    }
    __syncthreads();

#pragma unroll
    for (int k0 = 0; k0 < KPANEL; k0 += 4) {
      v2f a;
      a.x = ap[kk + k0];                    // A[m=lane16][kk+k0+khalf]
      a.y = ap[kk + k0 + 1];
      v2f b;
      b.x = wsp[k0 * WS_STRIDE];            // B[kk+k0+khalf][n=col0+lane16]
      b.y = wsp[(k0 + 1) * WS_STRIDE];
      // 8 args: (neg_a, A, neg_b, B, c_mod, C, reuse_a, reuse_b)
      c = __builtin_amdgcn_wmma_f32_16x16x4_f32(
          false, a, false, b, (short)0, c, false, false);
    }
  }

  // C/D layout: VGPR r -> M = r (lanes 0-15) or r+8 (lanes 16-31), N = lane%16
  const int n     = col0 + lane16;
  const int mbase = row0 + ((lane >> 4) << 3);
#pragma unroll
  for (int r = 0; r < 8; ++r)
    h[(size_t)(mbase + r) * GAT_OUT + n] = c[r];
}

// -------------------------------------------------------------------------
// Kernel 2: per-edge score s = relu(|h_i - h_j| . a), then 16-wide segment
// softmax in-register via 16-lane shuffle reductions (wave32). One thread
// per edge; 256 threads = 16 nodes per block; edges grouped 16 per
// destination row, 16-aligned (guaranteed by the input builder).
// -------------------------------------------------------------------------
__global__ __launch_bounds__(256) void sgl_attn_softmax(
    const float* __restrict__ h, const float* __restrict__ a,
    const int* __restrict__ rows, const int* __restrict__ cols,
    float* __restrict__ attn) {
  const int e = blockIdx.x * 256 + threadIdx.x;
  const int r = rows[e];
  const int cidx = cols[e];

  const float* hi = h + (size_t)r    * GAT_OUT;
  const float* hj = h + (size_t)cidx * GAT_OUT;

  float s = 0.f;
#pragma unroll
  for (int k = 0; k < GAT_OUT; k += 4) {
    float4 vi = *reinterpret_cast<const float4*>(hi + k);
    float4 vj = *reinterpret_cast<const float4*>(hj + k);
    float4 va = *reinterpret_cast<const float4*>(a + k);   // uniform -> s_loads
    s += fabsf(vi.x - vj.x) * va.x;
    s += fabsf(vi.y - vj.y) * va.y;
    s += fabsf(vi.z - vj.z) * va.z;
    s += fabsf(vi.w - vj.w) * va.w;
  }
  s = fmaxf(s, 0.f);   // relu

  // segment max over this node's 16 edges (lanes [0,15] / [16,31] groups)
  float m = s;
#pragma unroll
  for (int mask = 1; mask < GAT_DEG; mask <<= 1)
    m = fmaxf(m, __shfl_xor(m, mask, 32));

  float p = expf(s - m);

  float sum = p;
#pragma unroll
  for (int mask = 1; mask < GAT_DEG; mask <<= 1)
    sum += __shfl_xor(sum, mask, 32);

  attn[e] = p / sum;
}

// -------------------------------------------------------------------------
extern "C" void kernel_launch(void* const* d_in, const int* in_sizes, int n_in,
                              void* d_out, int out_size, void* d_ws, size_t ws_size,
                              hipStream_t stream) {
  const float* x  = (const float*)d_in[0];   // [N, 512]
  const float* W  = (const float*)d_in[1];   // [512, 128]
  const float* a  = (const float*)d_in[2];   // [128, 1]
  const int*   ei = (const int*)d_in[3];     // [2, E]

  const int N = in_sizes[0] / GAT_IN;        // 50000
  const int E = in_sizes[3] / 2;             // 800000

  float* h    = (float*)d_out;               // [N, 128]
  float* attn = h + (size_t)N * GAT_OUT;     // [E]

  sgl_gemm_h_f32wmma<<<N / 16, 256, 0, stream>>>(x, W, h);
  sgl_attn_softmax<<<E / 256, 256, 0, stream>>>(h, a, ei, ei + E, attn);
}